// GCN_3092376453801
// MI455X (gfx1250) — compile-verified
//
#include <hip/hip_runtime.h>
#include <hip/hip_bf16.h>
#include <cstdint>

typedef __attribute__((ext_vector_type(2))) float v2f;
typedef __attribute__((ext_vector_type(8))) float v8f;

#define D_IN 256
#define HID  128
#define NCLS 40
#define EPS  1e-6f

// ---------------------------------------------------------------------------
// Degree / normalization coefficient kernels
// ---------------------------------------------------------------------------
__global__ void deg_init_kernel(float* __restrict__ deg, int n) {
    int i = blockIdx.x * blockDim.x + threadIdx.x;
    if (i < n) deg[i] = 1.0f;   // self-loop contributes 1
}

__global__ void deg_accum_kernel(const int* __restrict__ dst, float* __restrict__ deg, int e) {
    int i = blockIdx.x * blockDim.x + threadIdx.x;
    if (i < e) atomicAdd(&deg[dst[i]], 1.0f);
}

__global__ void deg_inv_kernel(float* __restrict__ deg, int n) {
    int i = blockIdx.x * blockDim.x + threadIdx.x;
    if (i < n) deg[i] = rsqrtf(fmaxf(deg[i], 1.0f));
}

// ---------------------------------------------------------------------------
// Transpose + zero-pad weights: Bt[Mpad x K] from B[K x M]
// Bt[m*K + k] = (m < M) ? B[k*M + m] : 0
// ---------------------------------------------------------------------------
__global__ void transpose_pad_kernel(const float* __restrict__ B,
                                     float* __restrict__ Bt,
                                     int K, int M, int Mpad) {
    int idx = blockIdx.x * blockDim.x + threadIdx.x;
    int total = Mpad * K;
    if (idx >= total) return;
    int m = idx / K;
    int k = idx - m * K;
    Bt[idx] = (m < M) ? B[(size_t)k * M + m] : 0.0f;
}

// ---------------------------------------------------------------------------
// Dense GEMM: C[N x M] = A[N x K] * Bt^T, f32 WMMA 16x16x4.
// Bt is the padded transposed weight [Mpad x K]. One wave -> one 16x16 C tile.
// blockDim = (32, 4). No divergent predication in the K loop.
// ---------------------------------------------------------------------------
__global__ void gemm_wmma_f32(const float* __restrict__ A,
                              const float* __restrict__ Bt,
                              float* __restrict__ C,
                              int nrows, int K, int M) {
    const int lane = threadIdx.x;               // 0..31
    const int row0 = blockIdx.x * 16;
    const int col0 = (blockIdx.y * blockDim.y + threadIdx.y) * 16;
    if (col0 >= M) return;                      // wave-uniform: EXEC stays all-1

    const int l15   = lane & 15;
    const int khalf = (lane >> 4) << 1;         // 0 for lanes 0-15, 2 for 16-31
    const int arow  = row0 + l15;
    const int bcol  = col0 + l15;               // always < Mpad (zero-padded)

    const float* __restrict__ ap = A  + (size_t)arow * K + khalf;
    const float* __restrict__ bp = Bt + (size_t)bcol * K + khalf;

    v8f acc = {};
#pragma unroll 4
    for (int k0 = 0; k0 < K; k0 += 4) {
        const float2 av = *(const float2*)ap;  ap += 4;
        const float2 bv = *(const float2*)bp;  bp += 4;
        v2f a; a.x = av.x; a.y = av.y;
        v2f b; b.x = bv.x; b.y = bv.y;
        acc = __builtin_amdgcn_wmma_f32_16x16x4_f32(
            /*neg_a=*/false, a, /*neg_b=*/false, b,
            /*c_mod=*/(short)0, acc, /*reuse_a=*/false, /*reuse_b=*/false);
    }

    // C/D layout: VGPR r -> row = row0 + r + 8*(lane>=16), col = col0 + (lane&15)
    const int rbase = row0 + ((lane >> 4) << 3);
    const int c     = col0 + l15;
    if (c < M) {
#pragma unroll
        for (int r = 0; r < 8; ++r) {
            const int rr = rbase + r;
            if (rr < nrows) C[(size_t)rr * M + c] = acc[r];
        }
    }
}

// ---------------------------------------------------------------------------
// out[i,f] = h[i,f] * inv[i]^2 + b[f]   (self-loop term + bias, initializes out)
// ---------------------------------------------------------------------------
__global__ void self_bias_kernel(const float* __restrict__ h,
                                 const float* __restrict__ inv,
                                 const float* __restrict__ bias,
                                 float* __restrict__ out,
                                 int n, int F) {
    long long i = (long long)blockIdx.x * blockDim.x + threadIdx.x;
    long long total = (long long)n * F;
    if (i >= total) return;
    int node = (int)(i / F);
    int f    = (int)(i - (long long)node * F);
    float iv = inv[node];
    out[i] = h[i] * iv * iv + bias[f];
}

// ---------------------------------------------------------------------------
// Edge scatter: one wave per edge; lane handles 4 contiguous features.
// out[dst] += h[src] * inv[src]*inv[dst]
// ---------------------------------------------------------------------------
__global__ void mp_scatter_kernel(const float* __restrict__ h,
                                  const int* __restrict__ src,
                                  const int* __restrict__ dst,
                                  const float* __restrict__ inv,
                                  float* __restrict__ out,
                                  int E, int F) {
    const int lane = threadIdx.x & 31;
    const long long e = (long long)blockIdx.x * (blockDim.x >> 5) + (threadIdx.x >> 5);
    if (e >= E) return;
    const int s = src[e];
    const int d = dst[e];
    const float c = inv[s] * inv[d];
    const int f = lane << 2;
    if (f < F) {
        const float4 v = *(const float4*)(h + (size_t)s * F + f);
        float* op = out + (size_t)d * F + f;
        atomicAdd(op + 0, v.x * c);
        atomicAdd(op + 1, v.y * c);
        atomicAdd(op + 2, v.z * c);
        atomicAdd(op + 3, v.w * c);
    }
}

// ---------------------------------------------------------------------------
// PairNorm stats (F = 128 fixed): column sums + total sum of squares
// stats[0..127] = colsum, stats[128] = sumsq, stats[129] = scale
// ---------------------------------------------------------------------------
__global__ void zero_stats_kernel(float* __restrict__ stats) {
    int i = threadIdx.x;
    if (i < 130) stats[i] = 0.0f;
}

__global__ void stats_kernel(const float* __restrict__ x,
                             float* __restrict__ stats, int n) {
    __shared__ float ssq[128];
    const int f = threadIdx.x;                  // 0..127
    float cs = 0.0f, sq = 0.0f;
    for (int i = blockIdx.x; i < n; i += gridDim.x) {
        float v = x[(size_t)i * 128 + f];
        cs += v;
        sq += v * v;
    }
    atomicAdd(&stats[f], cs);
    ssq[f] = sq;
    __syncthreads();
#pragma unroll
    for (int s = 64; s > 0; s >>= 1) {
        if (f < s) ssq[f] += ssq[f + s];
        __syncthreads();
    }
    if (f == 0) atomicAdd(&stats[128], ssq[0]);
}

__global__ void finalize_stats_kernel(float* __restrict__ stats, float invN) {
    __shared__ float sm[128];
    const int f = threadIdx.x;
    float mean = stats[f] * invN;
    stats[f] = mean;                            // colsum -> mean in place
    sm[f] = mean * mean;
    __syncthreads();
#pragma unroll
    for (int s = 64; s > 0; s >>= 1) {
        if (f < s) sm[f] += sm[f + s];
        __syncthreads();
    }
    if (f == 0) {
        float ms = stats[128] * invN - sm[0];   // E[||xc||^2] per row
        stats[129] = rsqrtf(ms + EPS);
    }
}

__global__ void pairnorm_relu_kernel(float* __restrict__ x,
                                     const float* __restrict__ stats,
                                     long long total) {
    long long i = (long long)blockIdx.x * blockDim.x + threadIdx.x;
    if (i >= total) return;
    int f = (int)(i & 127);
    float v = (x[i] - stats[f]) * stats[129];
    x[i] = v > 0.0f ? v : 0.0f;
}

// ---------------------------------------------------------------------------
// Row-wise log_softmax over 40 classes
// ---------------------------------------------------------------------------
__global__ void log_softmax_kernel(const float* __restrict__ x,
                                   float* __restrict__ out, int n) {
    int i = blockIdx.x * blockDim.x + threadIdx.x;
    if (i >= n) return;
    const float* r = x + (size_t)i * NCLS;
    float m = r[0];
#pragma unroll
    for (int j = 1; j < NCLS; ++j) m = fmaxf(m, r[j]);
    float s = 0.0f;
#pragma unroll
    for (int j = 0; j < NCLS; ++j) s += expf(r[j] - m);
    float lse = logf(s);
    float* o = out + (size_t)i * NCLS;
#pragma unroll
    for (int j = 0; j < NCLS; ++j) o[j] = r[j] - m - lse;
}

// ---------------------------------------------------------------------------
// Launch
// ---------------------------------------------------------------------------
extern "C" void kernel_launch(void* const* d_in, const int* in_sizes, int n_in,
                              void* d_out, int out_size, void* d_ws, size_t ws_size,
                              hipStream_t stream) {
    const float* x  = (const float*)d_in[0];
    const int*   ei = (const int*)d_in[1];
    const float* W1 = (const float*)d_in[2];
    const float* b1 = (const float*)d_in[3];
    const float* W2 = (const float*)d_in[4];
    const float* b2 = (const float*)d_in[5];
    const float* W3 = (const float*)d_in[6];
    const float* b3 = (const float*)d_in[7];

    const int N = in_sizes[0] / D_IN;      // 100000
    const int E = in_sizes[1] / 2;         // 1600000
    const int* src = ei;
    const int* dst = ei + E;
    float* out = (float*)d_out;

    // Workspace layout
    float* inv   = (float*)d_ws;                       // N floats
    float* bufA  = inv  + (size_t)N;                   // N*HID floats
    float* bufB  = bufA + (size_t)N * HID;             // N*HID floats
    float* stats = bufB + (size_t)N * HID;             // 130 floats
    float* Bt    = stats + 256;                        // up to 128*256 floats

    const int T = 256;
    const int rowTiles = (N + 15) / 16;                // 6250
    const dim3 gblk(32, 4);
    const int NCLS_PAD = 48;                           // ceil16(40)

    // --- degree / inv sqrt ---
    deg_init_kernel<<<(N + T - 1) / T, T, 0, stream>>>(inv, N);
    deg_accum_kernel<<<(E + T - 1) / T, T, 0, stream>>>(dst, inv, E);
    deg_inv_kernel<<<(N + T - 1) / T, T, 0, stream>>>(inv, N);

    // ======================= Layer 1: X[N,256] @ W1 -> 128 ==================
    transpose_pad_kernel<<<(HID * D_IN + T - 1) / T, T, 0, stream>>>(W1, Bt, D_IN, HID, HID);
    gemm_wmma_f32<<<dim3(rowTiles, 2), gblk, 0, stream>>>(x, Bt, bufA, N, D_IN, HID);
    self_bias_kernel<<<(int)(((long long)N * HID + T - 1) / T), T, 0, stream>>>(
        bufA, inv, b1, bufB, N, HID);
    mp_scatter_kernel<<<(E + 7) / 8, T, 0, stream>>>(bufA, src, dst, inv, bufB, E, HID);
    zero_stats_kernel<<<1, 256, 0, stream>>>(stats);
    stats_kernel<<<512, 128, 0, stream>>>(bufB, stats, N);
    finalize_stats_kernel<<<1, 128, 0, stream>>>(stats, 1.0f / (float)N);
    pairnorm_relu_kernel<<<(int)(((long long)N * HID + T - 1) / T), T, 0, stream>>>(
        bufB, stats, (long long)N * HID);

    // ======================= Layer 2: H[N,128] @ W2 -> 128 ==================
    transpose_pad_kernel<<<(HID * HID + T - 1) / T, T, 0, stream>>>(W2, Bt, HID, HID, HID);
    gemm_wmma_f32<<<dim3(rowTiles, 2), gblk, 0, stream>>>(bufB, Bt, bufA, N, HID, HID);
    self_bias_kernel<<<(int)(((long long)N * HID + T - 1) / T), T, 0, stream>>>(
        bufA, inv, b2, bufB, N, HID);
    mp_scatter_kernel<<<(E + 7) / 8, T, 0, stream>>>(bufA, src, dst, inv, bufB, E, HID);
    zero_stats_kernel<<<1, 256, 0, stream>>>(stats);
    stats_kernel<<<512, 128, 0, stream>>>(bufB, stats, N);
    finalize_stats_kernel<<<1, 128, 0, stream>>>(stats, 1.0f / (float)N);
    pairnorm_relu_kernel<<<(int)(((long long)N * HID + T - 1) / T), T, 0, stream>>>(
        bufB, stats, (long long)N * HID);

    // ======================= Layer 3: H[N,128] @ W3 -> 40 ===================
    transpose_pad_kernel<<<(NCLS_PAD * HID + T - 1) / T, T, 0, stream>>>(W3, Bt, HID, NCLS, NCLS_PAD);
    gemm_wmma_f32<<<dim3(rowTiles, 1), gblk, 0, stream>>>(bufB, Bt, bufA, N, HID, NCLS);
    self_bias_kernel<<<(int)(((long long)N * NCLS + T - 1) / T), T, 0, stream>>>(
        bufA, inv, b3, bufB, N, NCLS);
    mp_scatter_kernel<<<(E + 7) / 8, T, 0, stream>>>(bufA, src, dst, inv, bufB, E, NCLS);

    // --- log_softmax -> d_out ---
    log_softmax_kernel<<<(N + T - 1) / T, T, 0, stream>>>(bufB, out, N);
}